// VGNN_56968446214866
// MI455X (gfx1250) — compile-verified
//
#include <hip/hip_runtime.h>

// Problem constants (from reference)
#define DD      64          // D
#define NC      18          // C categories
#define NP      8           // P prefs
#define DECAYC  0.1f

typedef __attribute__((ext_vector_type(2))) float v2f;
typedef __attribute__((ext_vector_type(8))) float v8f;

__device__ __forceinline__ float readlane_f(float x, int l) {
    return __builtin_bit_cast(float, __builtin_amdgcn_readlane(__builtin_bit_cast(int, x), l));
}

// One wave32 per edge.
// Per-edge math (algebraically reduced):
//   np_dot[p] = <news_pref[dst,cate,p,:64], src_emb>          (p = 0..7)
//   tg_dot    = <gnn_emb[dst,:64],           src_emb>
//   dot[p]    = w[p]*np_dot[p] + (1-w[p])*tg_dot,  w[p] = exp(-0.1*(t0 - last_update[p]))
//   logits    = dot * 1/sqrt(64);  attn = softmax(logits);  score = sum attn[p]*dot[p]
// The 9 length-64 dots go through V_WMMA_F32_16X16X4_F32 (fp32-exact path):
//   A(16x64): rows 0..7 = news rows, rows 8..15 = gnn row (8 needed, 9..15 harmless
//             finite filler — D rows 9..15 are never read).
//   B(64x16): every column = src_emb chunk (only column 0 is read; identical columns
//             keep all lanes' loads unconditional and finite).
// 16 K=4 steps, split over two accumulator chains to halve the dependent-WMMA latency.
// D[j,0] -> lane 0 acc[j] (j<8); D[8,0] -> lane 16 acc[0].
__global__ __launch_bounds__(256) void vgnn_edge_score_kernel(
    const float* __restrict__ src_rep,    // [N_src, D]
    const float* __restrict__ lastup,     // [N_dst, C, P]
    const float* __restrict__ newspref,   // [N_dst, C, P, 2D]
    const float* __restrict__ gnnemb,     // [N_dst, 2D]
    const float* __restrict__ timep,      // [E] (read [0])
    const int*   __restrict__ cate_id,    // [N_src]
    const int*   __restrict__ edge_src,   // [E]
    const int*   __restrict__ edge_dst,   // [E]
    float*       __restrict__ out,        // [E]
    int E)
{
    const int wave = __builtin_amdgcn_readfirstlane((int)(threadIdx.x >> 5));
    const int e = (int)blockIdx.x * 8 + wave;
    if (e >= E) return;                       // wave-uniform exit -> EXEC all-1s for WMMA

    const int lane = (int)(threadIdx.x & 31);
    const int r    = lane & 15;               // A row index (B column index is don't-care)
    const int hi   = lane >> 4;               // lane-half selects K pair {2hi, 2hi+1}

    // Wave-uniform gather indices (scalarized by the compiler)
    const int s  = edge_src[e];
    const int d  = edge_dst[e];
    const int c  = cate_id[s];

    const float* np_base = newspref + ((size_t)d * NC + (size_t)c) * (size_t)(NP * 2 * DD);
    const float* gnn_row = gnnemb   + (size_t)d * (2 * DD);
    const float* src_row = src_rep  + (size_t)s * DD;
    const float* lu_ptr  = lastup   + ((size_t)d * NC + (size_t)c) * NP;

    // Rows 0..7: news_pref rows; rows 8..15: gnn row (row 8 real, 9..15 filler)
    const float* aptr = (r < NP) ? (np_base + (size_t)r * (2 * DD)) : gnn_row;

    v2f a[16], b[16];
#pragma unroll
    for (int k = 0; k < 16; ++k) {
        const int off = 4 * k + 2 * hi;       // K-chunk k, this lane-half's K pair
        a[k] = *(const v2f*)(aptr + off);     // unconditional, always in-bounds
        b[k] = *(const v2f*)(src_row + off);  // same chunk in every B column
    }

    v8f acc0 = {}, acc1 = {};
#pragma unroll
    for (int k = 0; k < 16; k += 2) {
        acc0 = __builtin_amdgcn_wmma_f32_16x16x4_f32(
            false, a[k],     false, b[k],     (short)0, acc0, false, false);
        acc1 = __builtin_amdgcn_wmma_f32_16x16x4_f32(
            false, a[k + 1], false, b[k + 1], (short)0, acc1, false, false);
    }
    v8f acc;
#pragma unroll
    for (int j = 0; j < 8; ++j) acc[j] = acc0[j] + acc1[j];

    // Broadcast the 9 dot products to every lane (v_readlane)
    float npd[8];
#pragma unroll
    for (int j = 0; j < 8; ++j) npd[j] = readlane_f(acc[j], 0);   // D[j,0]
    const float tg = readlane_f(acc[0], 16);                      // D[8,0]

    const float t0 = timep[0];
    const float4 l0 = *(const float4*)(lu_ptr);       // 32B-aligned (index multiple of P=8)
    const float4 l1 = *(const float4*)(lu_ptr + 4);
    const float lu[8] = {l0.x, l0.y, l0.z, l0.w, l1.x, l1.y, l1.z, l1.w};

    float dotv[8], lg[8];
    float mx = -3.0e38f;
#pragma unroll
    for (int p = 0; p < 8; ++p) {
        const float w = __expf(-DECAYC * (t0 - lu[p]));
        dotv[p] = w * npd[p] + (1.f - w) * tg;
        lg[p]   = dotv[p] * 0.125f;                   // 1/sqrt(64)
        mx = fmaxf(mx, lg[p]);
    }
    float ssum = 0.f, sc = 0.f;
#pragma unroll
    for (int p = 0; p < 8; ++p) {
        const float ex = __expf(lg[p] - mx);
        ssum += ex;
        sc   += ex * dotv[p];
    }

    if (lane == 0) out[e] = sc / ssum;
}

extern "C" void kernel_launch(void* const* d_in, const int* in_sizes, int n_in,
                              void* d_out, int out_size, void* d_ws, size_t ws_size,
                              hipStream_t stream) {
    const float* src_rep  = (const float*)d_in[0];   // src_representation
    const float* lastup   = (const float*)d_in[1];   // dst_last_update
    const float* newspref = (const float*)d_in[2];   // dst_news_pref
    const float* gnnemb   = (const float*)d_in[3];   // dst_gnn_emb
    const float* timep    = (const float*)d_in[4];   // time
    const int*   cate_id  = (const int*)d_in[5];
    const int*   edge_src = (const int*)d_in[6];
    const int*   edge_dst = (const int*)d_in[7];

    const int E = in_sizes[6];                       // number of edges
    const int blocks = (E + 7) / 8;                  // 8 waves (edges) per 256-thread block
    vgnn_edge_score_kernel<<<blocks, 256, 0, stream>>>(
        src_rep, lastup, newspref, gnnemb, timep, cate_id, edge_src, edge_dst,
        (float*)d_out, E);
}